// FNO2d_18287970746604
// MI455X (gfx1250) — compile-verified
//
#include <hip/hip_runtime.h>
#include <math.h>

// ---------------- problem constants ----------------
enum { BB = 16, HH = 256, WW = 256, CH = 32, NLAYER = 4, MM1 = 12, MM2 = 12 };
enum { NPIX = BB * HH * WW };                 // 1,048,576 pixels
#define EPSV 1e-5f

// LDS padding strides (64 banks x 4B): stride mod 64 == 4 -> conflict-free
enum { T1S = 132, T2S = 68, TLS = 33 };

typedef __attribute__((ext_vector_type(2))) float v2f;
typedef __attribute__((ext_vector_type(8))) float v8f;

// V_WMMA_F32_16X16X4_F32 : D(16x16 f32) = A(16x4 f32) * B(4x16 f32) + C
static __device__ __forceinline__ v8f wmma4(v2f a, v2f b, v8f c) {
    return __builtin_amdgcn_wmma_f32_16x16x4_f32(false, a, false, b, (short)0, c,
                                                 false, false);
}

static __device__ __forceinline__ float gelu_erf(float x) {
    return 0.5f * x * (1.0f + erff(x * 0.70710678118654752f));
}

// ============================================================
// fc0 (3->32) + LayerNorm(32), channels-last output h[B,H,W,32]
// ============================================================
__global__ void k_fc0_ln(const float* __restrict__ x, const float* __restrict__ w,
                         const float* __restrict__ b, const float* __restrict__ g,
                         const float* __restrict__ bet, float* __restrict__ h) {
    int p = blockIdx.x * blockDim.x + threadIdx.x;
    if (p >= NPIX) return;
    float x0 = x[(size_t)p * 3 + 0];
    float x1 = x[(size_t)p * 3 + 1];
    float x2 = x[(size_t)p * 3 + 2];
    float v[CH];
    float mu = 0.f;
#pragma unroll
    for (int c = 0; c < CH; ++c) {
        float t = x0 * w[c] + x1 * w[CH + c] + x2 * w[2 * CH + c] + b[c];
        v[c] = t;
        mu += t;
    }
    mu *= (1.0f / CH);
    float var = 0.f;
#pragma unroll
    for (int c = 0; c < CH; ++c) {
        float d = v[c] - mu;
        var += d * d;
    }
    var *= (1.0f / CH);
    float rs = rsqrtf(var + EPSV);
    float* dst = h + (size_t)p * CH;
#pragma unroll
    for (int c = 0; c < CH; ++c) dst[c] = (v[c] - mu) * rs * g[c] + bet[c];
}

// ============================================================
// Partial forward DFT along W: Y[b,h,kx(12),c] = sum_w h * e^{-2πi w kx/256}
// one block per (b,h) row, 384 threads = (kx,c).
// Row staged into LDS with GLOBAL_LOAD_ASYNC_TO_LDS_B128 (ASYNCcnt path,
// bypasses VGPRs), overlapped with twiddle-table generation.
// ============================================================
__global__ void k_dftw(const float* __restrict__ h, float* __restrict__ Yr,
                       float* __restrict__ Yi) {
    __shared__ float row[WW * CH];  // 32 KB (one image row, all channels)
    __shared__ float ct[WW], st[WW];
    int bh = blockIdx.x;
    const float* src = h + (size_t)bh * WW * CH;

    // async bulk copy global -> LDS, 16B per lane per issue
    for (int i = threadIdx.x * 4; i < WW * CH; i += blockDim.x * 4) {
        unsigned ldsa = (unsigned)(size_t)(&row[i]);
        const float* ga = src + i;
        asm volatile("global_load_async_to_lds_b128 %0, %1, off"
                     :
                     : "v"(ldsa), "v"(ga)
                     : "memory");
    }
    // twiddle tables overlap the async copy
    for (int i = threadIdx.x; i < WW; i += blockDim.x) {
        float s, c;
        __sincosf(6.283185307179586f * (float)i * (1.0f / 256.0f), &s, &c);
        ct[i] = c;
        st[i] = s;
    }
    asm volatile("s_wait_asynccnt 0x0" ::: "memory");
    __syncthreads();

    int kx = threadIdx.x >> 5;
    int c = threadIdx.x & 31;
    float ar = 0.f, ai = 0.f;
    for (int w = 0; w < WW; ++w) {
        int idx = (w * kx) & 255;
        float v = row[w * CH + c];
        ar += v * ct[idx];
        ai -= v * st[idx];
    }
    size_t o = ((size_t)bh * MM2 + kx) * CH + c;
    Yr[o] = ar;
    Yi[o] = ai;
}

// ============================================================
// Partial forward DFT along H: Z[b,j(24),kx,c], j<12 -> ky=j, j>=12 -> ky=232+j
// grid (B,24), 384 threads = (kx,c)
// ============================================================
__global__ void k_dfth(const float* __restrict__ Yr, const float* __restrict__ Yi,
                       float* __restrict__ Zr, float* __restrict__ Zi) {
    __shared__ float ct[256], st[256];
    for (int i = threadIdx.x; i < 256; i += blockDim.x) {
        float s, c;
        __sincosf(6.283185307179586f * (float)i * (1.0f / 256.0f), &s, &c);
        ct[i] = c;
        st[i] = s;
    }
    __syncthreads();
    int b = blockIdx.x;
    int j = blockIdx.y;
    int kyv = (j < MM1) ? j : (232 + j);  // 244..255 for j = 12..23
    int kx = threadIdx.x >> 5;
    int c = threadIdx.x & 31;
    float zr = 0.f, zi = 0.f;
    for (int hh = 0; hh < HH; ++hh) {
        int idx = (hh * kyv) & 255;
        size_t o = (((size_t)b * HH + hh) * MM2 + kx) * CH + c;
        float yr = Yr[o], yi = Yi[o];
        float cc = ct[idx], ss = st[idx];
        zr += yr * cc + yi * ss;   // (yr+iyi)*(cc - i ss)
        zi += yi * cc - yr * ss;
    }
    size_t o2 = (((size_t)b * 24 + j) * MM2 + kx) * CH + c;
    Zr[o2] = zr;
    Zi[o2] = zi;
}

// ============================================================
// Spectral channel mix per mode: O[b,o] = sum_i Z[b,i] * W[i,o]  (complex)
// 288 modes, one wave per mode. WMMA f32: M=16(b), K=32(i), N=32(o).
// ============================================================
__global__ void k_spec(const float* __restrict__ Zr, const float* __restrict__ Zi,
                       const float* __restrict__ w1r, const float* __restrict__ w1i,
                       const float* __restrict__ w2r, const float* __restrict__ w2i,
                       float* __restrict__ Or, float* __restrict__ Oi) {
    int m = blockIdx.x;
    int bsel = m / 144;
    int rem = m % 144;
    int ky = rem / MM2;
    int kx = rem % MM2;
    int j = bsel * MM1 + ky;
    const float* Wr = bsel ? w2r : w1r;  // [i,o,ky,kx]
    const float* Wi = bsel ? w2i : w1i;

    int lane = threadIdx.x;
    int mrow = lane & 15;              // b index
    int khalf = (lane >> 4) << 1;      // 0 or 2
    size_t abase = (((size_t)mrow * 24 + j) * MM2 + kx) * CH;

    v8f aR[2] = {v8f{0, 0, 0, 0, 0, 0, 0, 0}, v8f{0, 0, 0, 0, 0, 0, 0, 0}};
    v8f aI[2] = {v8f{0, 0, 0, 0, 0, 0, 0, 0}, v8f{0, 0, 0, 0, 0, 0, 0, 0}};

    for (int kc = 0; kc < 8; ++kc) {
        int k0 = kc * 4 + khalf;
        v2f ar;
        ar.x = Zr[abase + k0];
        ar.y = Zr[abase + k0 + 1];
        v2f ai;
        ai.x = Zi[abase + k0];
        ai.y = Zi[abase + k0 + 1];
        v2f ain = -ai;  // for Re part: Zr*Wr - Zi*Wi
#pragma unroll
        for (int t = 0; t < 2; ++t) {
            int n = (lane & 15) + 16 * t;
            // W element (i=k, o=n): idx = ((k*32 + n)*12 + ky)*12 + kx
            int i0 = ((k0 * CH + n) * MM1 + ky) * MM2 + kx;
            int i1 = (((k0 + 1) * CH + n) * MM1 + ky) * MM2 + kx;
            v2f br;
            br.x = Wr[i0];
            br.y = Wr[i1];
            v2f bi;
            bi.x = Wi[i0];
            bi.y = Wi[i1];
            aR[t] = wmma4(ar, br, aR[t]);
            aR[t] = wmma4(ain, bi, aR[t]);
            aI[t] = wmma4(ar, bi, aI[t]);
            aI[t] = wmma4(ai, br, aI[t]);
        }
    }
    int mb = (lane >> 4) * 8;
#pragma unroll
    for (int t = 0; t < 2; ++t) {
        int n = (lane & 15) + 16 * t;
#pragma unroll
        for (int v = 0; v < 8; ++v) {
            size_t o = (((size_t)(mb + v) * 24 + j) * MM2 + kx) * CH + n;
            Or[o] = aR[t][v];
            Oi[o] = aI[t][v];
        }
    }
}

// ============================================================
// Inverse DFT along H for the 12 kx columns: F[b,h,kx,c] = sum_j Z2 * e^{+2πi h ky/256}
// grid B*H, 384 threads
// ============================================================
__global__ void k_idfth(const float* __restrict__ Zr, const float* __restrict__ Zi,
                        float* __restrict__ Fr, float* __restrict__ Fi) {
    __shared__ float ct[256], st[256];
    for (int i = threadIdx.x; i < 256; i += blockDim.x) {
        float s, c;
        __sincosf(6.283185307179586f * (float)i * (1.0f / 256.0f), &s, &c);
        ct[i] = c;
        st[i] = s;
    }
    __syncthreads();
    int b = blockIdx.x >> 8;
    int hh = blockIdx.x & 255;
    int kx = threadIdx.x >> 5;
    int c = threadIdx.x & 31;
    float fr = 0.f, fi = 0.f;
    for (int j = 0; j < 24; ++j) {
        int kyv = (j < MM1) ? j : (232 + j);
        int idx = (hh * kyv) & 255;
        size_t o = (((size_t)b * 24 + j) * MM2 + kx) * CH + c;
        float zr = Zr[o], zi = Zi[o];
        float cc = ct[idx], ss = st[idx];
        fr += zr * cc - zi * ss;  // (zr+izi)*(cc + i ss)
        fi += zr * ss + zi * cc;
    }
    size_t o2 = ((size_t)blockIdx.x * MM2 + kx) * CH + c;
    Fr[o2] = fr;
    Fi[o2] = fi;
}

// ============================================================
// Fused: irfft along W (12 modes) + pointwise conv (WMMA, 32x32) + residual
//        + LayerNorm + optional gelu, updating h IN PLACE.
// grid B*H*2, 256 threads (8 waves x 16 pixels = 128 pixels per block)
// ============================================================
__global__ void k_conv_ln(const float* __restrict__ Fr, const float* __restrict__ Fi,
                          const float* __restrict__ cw, const float* __restrict__ cb,
                          const float* __restrict__ g, const float* __restrict__ bet,
                          float* __restrict__ h, int do_gelu) {
    __shared__ float sFr[MM2 * CH];
    __shared__ float sFi[MM2 * CH];
    __shared__ float ct[256], st[256];
    __shared__ float tile[128 * TLS];  // padded rows: x1 then (x1+x2+bias)

    int blk = blockIdx.x;
    int half = blk & 1;
    int bh = blk >> 1;
    int w0 = half * 128;

    for (int i = threadIdx.x; i < MM2 * CH; i += 256) {
        sFr[i] = Fr[(size_t)bh * MM2 * CH + i];
        sFi[i] = Fi[(size_t)bh * MM2 * CH + i];
    }
    for (int i = threadIdx.x; i < 256; i += 256) {
        float s, c;
        __sincosf(6.283185307179586f * (float)i * (1.0f / 256.0f), &s, &c);
        ct[i] = c;
        st[i] = s;
    }
    __syncthreads();

    // phase 1: x1 for 128 px x 32 ch (Hermitian-weighted inverse rDFT along W)
    for (int i = threadIdx.x; i < 128 * CH; i += 256) {
        int px = i >> 5;
        int c = i & 31;
        int w = w0 + px;
        float acc = sFr[c];  // kx = 0 term (DC along W, imag implied zero)
        for (int kx = 1; kx < MM2; ++kx) {
            int idx = (w * kx) & 255;
            acc += 2.0f * (sFr[kx * CH + c] * ct[idx] - sFi[kx * CH + c] * st[idx]);
        }
        tile[px * TLS + c] = acc * (1.0f / 65536.0f);  // 1/(H*W)
    }
    __syncthreads();

    // phase 2: WMMA pointwise conv: x2[px,o] = sum_c h[px,c] * cw[o,c]
    int lane = threadIdx.x & 31;
    int wave = threadIdx.x >> 5;
    const float* A = h + ((size_t)bh * WW + w0 + wave * 16) * CH;
    int khalf = (lane >> 4) << 1;
    v8f acc[2] = {v8f{0, 0, 0, 0, 0, 0, 0, 0}, v8f{0, 0, 0, 0, 0, 0, 0, 0}};
    for (int kc = 0; kc < 8; ++kc) {
        int k0 = kc * 4 + khalf;
        v2f a;
        a.x = A[(lane & 15) * CH + k0];
        a.y = A[(lane & 15) * CH + k0 + 1];
#pragma unroll
        for (int t = 0; t < 2; ++t) {
            int n = (lane & 15) + 16 * t;
            v2f b;  // cw is [o,c] -> transposed read
            b.x = cw[n * CH + k0];
            b.y = cw[n * CH + k0 + 1];
            acc[t] = wmma4(a, b, acc[t]);
        }
    }
    int mb = (lane >> 4) * 8;
#pragma unroll
    for (int t = 0; t < 2; ++t) {
        int n = (lane & 15) + 16 * t;
        float bn = cb[n];
#pragma unroll
        for (int v = 0; v < 8; ++v) {
            int idx = (wave * 16 + mb + v) * TLS + n;
            tile[idx] = tile[idx] + acc[t][v] + bn;  // x1 + x2 + bias
        }
    }
    __syncthreads();

    // phase 3: LayerNorm(32) + optional gelu; write back in place
    if (threadIdx.x < 128) {
        int px = threadIdx.x;
        const float* tp = &tile[px * TLS];
        float mu = 0.f;
#pragma unroll
        for (int c = 0; c < CH; ++c) mu += tp[c];
        mu *= (1.0f / CH);
        float var = 0.f;
#pragma unroll
        for (int c = 0; c < CH; ++c) {
            float d = tp[c] - mu;
            var += d * d;
        }
        var *= (1.0f / CH);
        float rs = rsqrtf(var + EPSV);
        float* dst = h + ((size_t)bh * WW + w0 + px) * CH;
#pragma unroll
        for (int c = 0; c < CH; ++c) {
            float y = (tp[c] - mu) * rs * g[c] + bet[c];
            if (do_gelu) y = gelu_erf(y);
            dst[c] = y;
        }
    }
}

// ============================================================
// Fused MLP head: gelu(h@fc1) -> gelu(@fc2) -> @fc3, per 16-pixel WMMA tile.
// grid NPIX/64, 128 threads (4 waves x 16 px). LDS rows padded (stride%64==4).
// ============================================================
__global__ void k_head(const float* __restrict__ h, const float* __restrict__ w1,
                       const float* __restrict__ b1, const float* __restrict__ w2,
                       const float* __restrict__ b2, const float* __restrict__ w3,
                       const float* __restrict__ b3, float* __restrict__ out) {
    __shared__ float t1[4][16 * T1S];  // ~33 KB
    __shared__ float t2[4][16 * T2S];  // ~17 KB
    int lane = threadIdx.x & 31;
    int wave = threadIdx.x >> 5;
    size_t px0 = (size_t)blockIdx.x * 64 + wave * 16;
    const float* A = h + px0 * CH;
    int khalf = (lane >> 4) << 1;
    int nl = lane & 15;
    int mb = (lane >> 4) * 8;

    // fc1: [16,32] @ [32,128], gelu -> t1
#pragma unroll
    for (int t = 0; t < 8; ++t) {
        v8f acc = v8f{0, 0, 0, 0, 0, 0, 0, 0};
        int n = nl + 16 * t;
        for (int kc = 0; kc < 8; ++kc) {
            int k0 = kc * 4 + khalf;
            v2f a;
            a.x = A[nl * CH + k0];
            a.y = A[nl * CH + k0 + 1];
            v2f b;  // w1 is [K=32, N=128] row-major
            b.x = w1[k0 * 128 + n];
            b.y = w1[(k0 + 1) * 128 + n];
            acc = wmma4(a, b, acc);
        }
        float bn = b1[n];
#pragma unroll
        for (int v = 0; v < 8; ++v) t1[wave][(mb + v) * T1S + n] = gelu_erf(acc[v] + bn);
    }
    __syncthreads();

    // fc2: [16,128] @ [128,64], gelu -> t2
#pragma unroll
    for (int t = 0; t < 4; ++t) {
        v8f acc = v8f{0, 0, 0, 0, 0, 0, 0, 0};
        int n = nl + 16 * t;
        for (int kc = 0; kc < 32; ++kc) {
            int k0 = kc * 4 + khalf;
            v2f a;
            a.x = t1[wave][nl * T1S + k0];
            a.y = t1[wave][nl * T1S + k0 + 1];
            v2f b;  // w2 is [K=128, N=64]
            b.x = w2[k0 * 64 + n];
            b.y = w2[(k0 + 1) * 64 + n];
            acc = wmma4(a, b, acc);
        }
        float bn = b2[n];
#pragma unroll
        for (int v = 0; v < 8; ++v) t2[wave][(mb + v) * T2S + n] = gelu_erf(acc[v] + bn);
    }
    __syncthreads();

    // fc3: dot over 64 channels, one pixel per lane (lanes 0..15)
    if (lane < 16) {
        float s = b3[0];
        for (int c = 0; c < 64; ++c) s += t2[wave][lane * T2S + c] * w3[c];
        out[px0 + lane] = s;
    }
}

// ============================================================
extern "C" void kernel_launch(void* const* d_in, const int* in_sizes, int n_in,
                              void* d_out, int out_size, void* d_ws, size_t ws_size,
                              hipStream_t stream) {
    (void)in_sizes;
    (void)n_in;
    (void)out_size;
    (void)ws_size;
    const float* x = (const float*)d_in[0];
    const float* fc0_w = (const float*)d_in[1];
    const float* fc0_b = (const float*)d_in[2];
    const float* ln0_g = (const float*)d_in[3];
    const float* ln0_b = (const float*)d_in[4];
    const float* w1r = (const float*)d_in[5];
    const float* w1i = (const float*)d_in[6];
    const float* w2r = (const float*)d_in[7];
    const float* w2i = (const float*)d_in[8];
    const float* conv_w = (const float*)d_in[9];
    const float* conv_b = (const float*)d_in[10];
    const float* ln_g = (const float*)d_in[11];
    const float* ln_b = (const float*)d_in[12];
    const float* fc1_w = (const float*)d_in[13];
    const float* fc1_b = (const float*)d_in[14];
    const float* fc2_w = (const float*)d_in[15];
    const float* fc2_b = (const float*)d_in[16];
    const float* fc3_w = (const float*)d_in[17];
    const float* fc3_b = (const float*)d_in[18];

    // workspace carve (floats): h | Yr | Yi | Fr | Fi | Zr | Zi | Z2r | Z2i
    const size_t nH = (size_t)NPIX * CH;           // 33.5M floats
    const size_t nY = (size_t)BB * HH * MM2 * CH;  // 1.57M floats
    const size_t nZ = (size_t)BB * 24 * MM2 * CH;  // 0.15M floats
    float* h = (float*)d_ws;
    float* Yr = h + nH;
    float* Yi = Yr + nY;
    float* Fr = Yi + nY;
    float* Fi = Fr + nY;
    float* Zr = Fi + nY;
    float* Zi = Zr + nZ;
    float* Z2r = Zi + nZ;
    float* Z2i = Z2r + nZ;

    k_fc0_ln<<<NPIX / 256, 256, 0, stream>>>(x, fc0_w, fc0_b, ln0_g, ln0_b, h);

    const int WOFF = CH * CH * MM1 * MM2;  // per-layer complex-weight stride
    for (int l = 0; l < NLAYER; ++l) {
        k_dftw<<<BB * HH, 384, 0, stream>>>(h, Yr, Yi);
        k_dfth<<<dim3(BB, 24), 384, 0, stream>>>(Yr, Yi, Zr, Zi);
        k_spec<<<288, 32, 0, stream>>>(Zr, Zi, w1r + (size_t)l * WOFF,
                                       w1i + (size_t)l * WOFF, w2r + (size_t)l * WOFF,
                                       w2i + (size_t)l * WOFF, Z2r, Z2i);
        k_idfth<<<BB * HH, 384, 0, stream>>>(Z2r, Z2i, Fr, Fi);
        k_conv_ln<<<BB * HH * 2, 256, 0, stream>>>(
            Fr, Fi, conv_w + (size_t)l * CH * CH, conv_b + (size_t)l * CH,
            ln_g + (size_t)l * CH, ln_b + (size_t)l * CH, h, (l < NLAYER - 1) ? 1 : 0);
    }

    k_head<<<NPIX / 64, 128, 0, stream>>>(h, fc1_w, fc1_b, fc2_w, fc2_b, fc3_w, fc3_b,
                                          (float*)d_out);
}